// RuleGraphConvLayer_78271484002763
// MI455X (gfx1250) — compile-verified
//
#include <hip/hip_runtime.h>

typedef float v2f __attribute__((ext_vector_type(2)));
typedef float v8f __attribute__((ext_vector_type(8)));

#define NATOMS 500000
#define FDIM 81
#define CDIM 128
#define KDIM 162   // stacked [w_s ; w_n] rows
#define KP   164   // padded K (multiple of 4)
#define WPB  3     // waves per block (wave32)
#define MROWS 32   // atoms per wave (two 16-row WMMA tiles share B frags)

__global__ __launch_bounds__(WPB * 32) void rulegc_kernel(
    const float* __restrict__ A, const int* __restrict__ nbr,
    const float* __restrict__ w_s, const float* __restrict__ w_n,
    float* __restrict__ out, int ntiles)
{
  // Transposed stacked weights: sW[c*KP + r], rows 162..163 zeroed. (82 KB)
  __shared__ __align__(16) float sW[CDIM * KP];
  // Per-wave X tile: 32 atoms x 164 cols ([self(81) | g(81) | pad(2)]) (3 x 21 KB)
  __shared__ __align__(16) float sX[WPB][MROWS * KP];

  const int tid  = threadIdx.x;
  const int lane = tid & 31;
  const int wv   = tid >> 5;
  const int half = lane >> 4;   // 0: lanes 0-15, 1: lanes 16-31
  const int r15  = lane & 15;

  // ---- Stage weights into LDS (coalesced global reads, transposed write) ----
  for (int i = tid; i < KP * CDIM; i += WPB * 32) {
    int r = i >> 7;           // 0..163 (row of stacked W)
    int c = i & (CDIM - 1);   // 0..127 (output channel)
    float v = 0.0f;
    if (r < FDIM)       v = w_s[r * CDIM + c];
    else if (r < KDIM)  v = w_n[(r - FDIM) * CDIM + c];
    sW[c * KP + r] = v;
  }

  const int gw = blockIdx.x * WPB + wv;     // global wave id
  const bool active = (gw * 2 < ntiles);    // wave owns tiles 2gw, 2gw+1
  const int a0 = gw * MROWS;

  if (active) {
    // Per-lane neighbor precompute, two sets of 16 atoms:
    //   set A: atoms a0 + (lane&15),      neighbor k = lane>>4
    //   set B: atoms a0 + 16 + (lane&15), neighbor k = lane>>4
    float scS[2];
    int   icS[2];
#pragma unroll
    for (int h = 0; h < 2; ++h) {
      int aq   = a0 + h * 16 + r15;
      int nidx = nbr[aq * 2 + half];
      bool valid = (nidx > 0) && (nidx < NATOMS);
      int ic = nidx < 0 ? 0 : (nidx > NATOMS - 1 ? NATOMS - 1 : nidx);
      const float* sp = A + (long)aq * FDIM;
      const float* np = A + (long)ic * FDIM;
      float dx = sp[0] - np[0];
      float dy = sp[1] - np[1];
      float dz = sp[2] - np[2];
      float d2 = dx * dx + dy * dy + dz * dz;
      // dist>0 <=> d2>0 ; 1/max(dist,1e-3)^2 == 1/max(d2,1e-6)
      float sc = (d2 > 0.0f) ? (1.0f / fmaxf(d2, 1e-6f)) : 1.0f;
      scS[h] = valid ? sc : 0.0f;  // fold valid-mask into the scale
      icS[h] = ic;
    }

    float* Xw = &sX[wv][0];
    for (int a = 0; a < MROWS; ++a) {
      const int h  = a >> 4;        // which precompute set (uniform)
      const int al = a & 15;
      float s0 = __shfl(scS[h], al, 32);
      float s1 = __shfl(scS[h], al + 16, 32);
      int   i0 = __shfl(icS[h], al, 32);
      int   i1 = __shfl(icS[h], al + 16, 32);
      const float* selfp = A + (long)(a0 + a) * FDIM;
      const float* n0p   = A + (long)i0 * FDIM;
      const float* n1p   = A + (long)i1 * FDIM;
      for (int c = lane; c < KP; c += 32) {
        float val;
        if (c < FDIM) {
          val = selfp[c];                       // self-features -> w_s path
        } else if (c < KDIM) {
          int f = c - FDIM;
          float s = (f >= 3) ? selfp[f] : 0.0f; // coords: part1 = neigh only
          float u = n0p[f];
          float v = n1p[f];
          val = (s + u) * s0 + (s + v) * s1;    // g = sum over both neighbors
        } else {
          val = 0.0f;                           // K padding
        }
        Xw[a * KP + c] = val;
      }
    }
  }

  __syncthreads();
  if (!active) return;

  // ---- GEMM: 32x164 (X) @ 164x128 (W) via V_WMMA_F32_16X16X4_F32 ----
  // Two M-tiles share every B fragment: 10 b64 DS loads per 16 WMMAs.
  v8f acc0[8] = {};
  v8f acc1[8] = {};
  const float* Xw = &sX[wv][0];
  for (int k0 = 0; k0 < KP; k0 += 4) {
    const int ca = k0 + 2 * half;  // A frag: lanes 0-15 K=k0,k0+1; 16-31 K=k0+2,k0+3
    v2f afA = *(const v2f*)&Xw[r15 * KP + ca];
    v2f afB = *(const v2f*)&Xw[(16 + r15) * KP + ca];
#pragma unroll
    for (int nt = 0; nt < 8; ++nt) {
      const int cn = nt * 16 + r15;  // B frag: N = lane&15 within 16-col tile
      v2f bf = *(const v2f*)&sW[cn * KP + ca];  // contiguous b64 (transposed W)
      acc0[nt] = __builtin_amdgcn_wmma_f32_16x16x4_f32(
          false, afA, false, bf, (short)0, acc0[nt], false, false);
      acc1[nt] = __builtin_amdgcn_wmma_f32_16x16x4_f32(
          false, afB, false, bf, (short)0, acc1[nt], false, false);
    }
  }

  // ---- Store: C/D layout: VGPR j -> row (j + 8*half), lane&15 -> column ----
  float* op0 = out + (long)(a0 + 8 * half) * CDIM + r15;
  float* op1 = out + (long)(a0 + 16 + 8 * half) * CDIM + r15;
#pragma unroll
  for (int nt = 0; nt < 8; ++nt) {
#pragma unroll
    for (int j = 0; j < 8; ++j) {
      op0[(long)j * CDIM + nt * 16] = acc0[nt][j];
      op1[(long)j * CDIM + nt * 16] = acc1[nt][j];
    }
  }
}

extern "C" void kernel_launch(void* const* d_in, const int* in_sizes, int n_in,
                              void* d_out, int out_size, void* d_ws, size_t ws_size,
                              hipStream_t stream) {
  const float* A   = (const float*)d_in[0];
  const int*   nbr = (const int*)d_in[1];
  const float* w_s = (const float*)d_in[2];
  const float* w_n = (const float*)d_in[3];
  float* out = (float*)d_out;
  int ntiles = (NATOMS + 15) / 16;               // 31250 (exact)
  int nwaves = (ntiles + 1) / 2;                 // 15625 waves, 32 atoms each
  int blocks = (nwaves + WPB - 1) / WPB;         // 5209
  rulegc_kernel<<<blocks, WPB * 32, 0, stream>>>(A, nbr, w_s, w_n, out, ntiles);
}